// HeatmapOffsetmapLoss_8375186227400
// MI455X (gfx1250) — compile-verified
//
#include <hip/hip_runtime.h>
#include <hip/hip_bf16.h>
#include <cstdint>

// Problem constants (from the reference).
constexpr int B_ = 2, L_ = 19, H_ = 800, W_ = 640;
constexpr int NP      = B_ * L_;          // 38 (b,l) pairs
constexpr int PLANE   = H_ * W_;          // 512000 pixels per channel plane
constexpr int NCHUNK  = 125;              // chunks per plane
constexpr int CHUNK   = PLANE / NCHUNK;   // 4096 pixels per chunk
constexpr int NTHR    = 256;              // 8 wave32 waves
constexpr int VEC     = 4;                // float4 per lane per stage
constexpr int STAGEPX = NTHR * VEC;       // 1024 pixels per stage
constexpr int NSTAGE  = CHUNK / STAGEPX;  // 4 stages per chunk
constexpr float HEAT_R2 = 40.0f * 40.0f;  // exact integer compare vs dist^2
constexpr float INV_R   = 1.0f / 40.0f;   // off-map scale

// ---- CDNA5 async global->LDS path (ASYNCcnt), non-temporal stream ----------
__device__ __forceinline__ void async_copy16(uint32_t ldsOff, const float* gptr) {
    // per-lane DMA of 16 bytes from global gptr to LDS offset ldsOff; streamed once -> NT hint
    asm volatile("global_load_async_to_lds_b128 %0, %1, off th:TH_LOAD_NT"
                 :
                 : "v"(ldsOff), "v"(gptr)
                 : "memory");
}
__device__ __forceinline__ void wait_async_le3() {
    asm volatile("s_wait_asynccnt 0x3" ::: "memory");
}
__device__ __forceinline__ void wait_async_0() {
    asm volatile("s_wait_asynccnt 0x0" ::: "memory");
}
__device__ __forceinline__ void wait_ds_0() {
    asm volatile("s_wait_dscnt 0x0" ::: "memory");
}
__device__ __forceinline__ uint32_t lds_off(const void* p) {
    // the low 32 bits of the flat shared pointer are the LDS byte offset
    return (uint32_t)(uintptr_t)p;
}

__device__ __forceinline__ float wave_sum(float v) {
    #pragma unroll
    for (int o = 16; o > 0; o >>= 1) v += __shfl_xor(v, o, 32);
    return v;
}

// softplus(-|z|) = log(1 + exp(-|z|)) via hardware v_exp_f32 / v_log_f32
__device__ __forceinline__ float softplus_neg_abs(float az) {
    return __logf(1.0f + __expf(-az));
}

// ---- Main streaming kernel: one (pair, chunk) per block ---------------------
__global__ __launch_bounds__(NTHR)
void heatmap_loss_partials(const float* __restrict__ fm,
                           const float* __restrict__ lms,
                           float4* __restrict__ part) {
    __shared__ float4 buf[3][2][NTHR];   // plane x double-buffer x lane, 24 KB
    __shared__ float4 red[NTHR / 32];

    const int tid   = threadIdx.x;
    const int p     = blockIdx.y;        // pair index 0..37
    const int chunk = blockIdx.x;        // 0..124
    const int bb    = p / L_;
    const int ll    = p % L_;

    // landmark -> integer-valued center (exact floats)
    const float x = floorf(lms[p * 2 + 0] * (float)H_);
    const float y = floorf(lms[p * 2 + 1] * (float)W_);

    const size_t base = (size_t)(bb * 3 * L_ + ll) * (size_t)PLANE;
    const float* zp = fm + base;                            // channel l
    const float* ap = fm + base + (size_t)L_ * PLANE;       // channel L+l
    const float* op = fm + base + (size_t)(2 * L_) * PLANE; // channel 2L+l

    const int q0 = chunk * CHUNK + tid * VEC;

    float bce_s = 0.f, l1a_s = 0.f, l1b_s = 0.f, cnt_s = 0.f;

    auto issue = [&](int s) {
        const int q  = q0 + s * STAGEPX;
        const int bi = s & 1;
        wait_ds_0();  // buffer reuse: ensure prior ds reads of this buffer done
        async_copy16(lds_off(&buf[0][bi][tid]), zp + q);
        async_copy16(lds_off(&buf[1][bi][tid]), ap + q);
        async_copy16(lds_off(&buf[2][bi][tid]), op + q);
    };

    issue(0);
    #pragma unroll
    for (int s = 0; s < NSTAGE; ++s) {
        if (s + 1 < NSTAGE) { issue(s + 1); wait_async_le3(); }
        else                { wait_async_0(); }

        const int bi = s & 1;
        float4 zv = buf[0][bi][tid];
        float4 av = buf[1][bi][tid];
        float4 ov = buf[2][bi][tid];

        const int   q    = q0 + s * STAGEPX;     // W%4==0 -> 4 pixels same row
        const float fi   = (float)(q / W_);
        const float fj0  = (float)(q % W_);
        const float dx   = fi - x;
        const float dx2  = dx * dx;
        const float off1 = ((float)(H_ - 1) + x - fi) * INV_R;

        const float* ze = &zv.x;
        const float* ae = &av.x;
        const float* oe = &ov.x;
        #pragma unroll
        for (int k = 0; k < VEC; ++k) {
            const float fj   = fj0 + (float)k;
            const float dy   = fj - y;
            const float heat = (dx2 + dy * dy <= HEAT_R2) ? 1.0f : 0.0f;
            const float z    = ze[k];
            bce_s += fmaxf(z, 0.0f) - z * heat + softplus_neg_abs(fabsf(z));
            const float off0 = ((float)(W_ - 1) + y - fj) * INV_R;
            l1a_s += fabsf(ae[k] - off1) * heat;
            l1b_s += fabsf(oe[k] - off0) * heat;
            cnt_s += heat;
        }
    }

    // deterministic block reduction: wave shuffles -> LDS -> wave0
    bce_s = wave_sum(bce_s);
    l1a_s = wave_sum(l1a_s);
    l1b_s = wave_sum(l1b_s);
    cnt_s = wave_sum(cnt_s);

    const int wave = tid >> 5, lane = tid & 31;
    if (lane == 0) red[wave] = make_float4(bce_s, l1a_s, l1b_s, cnt_s);
    __syncthreads();
    if (wave == 0) {
        float4 v = (lane < NTHR / 32) ? red[lane] : make_float4(0.f, 0.f, 0.f, 0.f);
        v.x = wave_sum(v.x); v.y = wave_sum(v.y);
        v.z = wave_sum(v.z); v.w = wave_sum(v.w);
        if (lane == 0) part[p * NCHUNK + chunk] = v;
    }
}

// ---- Finalize: single block, fixed-order reduction over 38x125 partials -----
__global__ __launch_bounds__(128)
void heatmap_loss_finalize(const float4* __restrict__ part,
                           float* __restrict__ out) {
    __shared__ float4 red[4];
    const int tid = threadIdx.x, wave = tid >> 5, lane = tid & 31;
    float loss = 0.f;

    for (int p = 0; p < NP; ++p) {
        float4 v = (tid < NCHUNK) ? part[p * NCHUNK + tid]
                                  : make_float4(0.f, 0.f, 0.f, 0.f);
        v.x = wave_sum(v.x); v.y = wave_sum(v.y);
        v.z = wave_sum(v.z); v.w = wave_sum(v.w);
        if (lane == 0) red[wave] = v;
        __syncthreads();
        if (tid == 0) {
            float4 t = red[0];
            #pragma unroll
            for (int w = 1; w < 4; ++w) {
                t.x += red[w].x; t.y += red[w].y;
                t.z += red[w].z; t.w += red[w].w;
            }
            loss += 2.0f * t.x / (float)PLANE + t.y / t.w + t.z / t.w;
        }
        __syncthreads();
    }
    if (tid == 0) out[0] = loss / (float)NP;
}

extern "C" void kernel_launch(void* const* d_in, const int* in_sizes, int n_in,
                              void* d_out, int out_size, void* d_ws, size_t ws_size,
                              hipStream_t stream) {
    const float* fm   = (const float*)d_in[0];  // (2, 57, 800, 640) fp32
    const float* lms  = (const float*)d_in[1];  // (2, 19, 2) fp32
    float*       out  = (float*)d_out;          // scalar fp32
    float4*      part = (float4*)d_ws;          // 38*125 float4 = 76 KB scratch

    dim3 grid(NCHUNK, NP);
    heatmap_loss_partials<<<grid, NTHR, 0, stream>>>(fm, lms, part);
    heatmap_loss_finalize<<<1, 128, 0, stream>>>(part, out);
}